// AttentionSampleUpdater_35708358099444
// MI455X (gfx1250) — compile-verified
//
#include <hip/hip_runtime.h>

// CDNA5 / gfx1250, wave32.
typedef __attribute__((ext_vector_type(2))) float v2f;
typedef __attribute__((ext_vector_type(8))) float v8f;

#define NEG_INF (-__builtin_inff())

static constexpr int Dm  = 32;   // embedding dim
static constexpr int Kk  = 8;    // samples per node
static constexpr int Dn  = 8;    // neighbors per node
static constexpr int P   = 72;   // D*K + K candidates
static constexpr int WPB = 8;    // waves (nodes) per 256-thread block

struct WaveScratch {
  int   cand[P];
  int   scand[P];
  float sims[80];    // 5 WMMA tiles x 16
  int   uniqf[P];
  int   topk[Kk];
  int   fb[Kk];
};

__global__ __launch_bounds__(256)
void AttentionSampleUpdater_kernel(const float* __restrict__ x,
                                   const int* __restrict__ neighbors,
                                   const int* __restrict__ samples,
                                   int* __restrict__ out, int N)
{
  __shared__ WaveScratch ws[WPB];
  const int wave = threadIdx.x >> 5;
  const int lane = threadIdx.x & 31;
  WaveScratch& S = ws[wave];

  const int  nodeRaw = blockIdx.x * WPB + wave;
  const bool validNode = nodeRaw < N;
  const int  node = validNode ? nodeRaw : (N - 1); // clamp: keep EXEC full for WMMA

  // ---- 1) gather candidate multiset: [neighbor samples (64) | own samples (8)]
  for (int j = lane; j < P; j += 32) {
    int v;
    if (j < Dn * Kk) {
      int nb = neighbors[node * Dn + (j >> 3)];
      v = samples[nb * Kk + (j & 7)];
    } else {
      v = samples[node * Kk + (j - Dn * Kk)];
    }
    S.cand[j] = v;
  }
  __syncthreads();

  // ---- 2) stable rank-sort of 72 ints (O(P^2) spread over 32 lanes)
  int myJ[3], myV[3], myR[3];
  int nslots = 0;
  for (int j = lane; j < P; j += 32) {
    myJ[nslots] = j; myV[nslots] = S.cand[j]; myR[nslots] = 0; ++nslots;
  }
  for (int k = 0; k < P; ++k) {
    int c = S.cand[k];                       // LDS broadcast (uniform addr)
    for (int t = 0; t < nslots; ++t)
      myR[t] += (c < myV[t]) || (c == myV[t] && k < myJ[t]);
  }
  for (int t = 0; t < nslots; ++t) S.scand[myR[t]] = myV[t];
  __syncthreads();

  // ---- 3) uniqueness flags (first occurrence in sorted order) + wave count
  int uf[3]; int num_u = 0;
  for (int t = 0; t < nslots; ++t) {
    int j = myJ[t];
    int u = (j == 0) || (S.scand[j] != S.scand[j - 1]);
    uf[t] = u; S.uniqf[j] = u; num_u += u;
  }
  for (int off = 16; off > 0; off >>= 1) num_u += __shfl_xor(num_u, off, 32);

  // ---- 4) sims via V_WMMA_F32_16X16X4_F32 chains (f32-exact dot products)
  // A (16x32): rows = candidate embeddings; lane = row M (mod 16),
  //            lane-half selects K pair within each 16x4 step.
  // B (32x16): all 16 columns = x[node]  -> D[m][*] = cand_m . x_i
  const int h = lane >> 4;
  const float* xi = x + (size_t)node * Dm;
  float2 bq[8];
  #pragma unroll
  for (int s = 0; s < 8; ++s)
    bq[s] = *(const float2*)(xi + 4 * s + 2 * h);

  const int m = lane & 15;
  #pragma unroll
  for (int tile = 0; tile < 5; ++tile) {
    int slot = tile * 16 + m;
    int c = (slot < P) ? S.scand[slot] : 0;   // pad rows read x[0]; masked later
    const float* xc = x + (size_t)c * Dm;
    v8f acc = {0.f, 0.f, 0.f, 0.f, 0.f, 0.f, 0.f, 0.f};
    #pragma unroll
    for (int s = 0; s < 8; ++s) {
      float2 af = *(const float2*)(xc + 4 * s + 2 * h);
      v2f A; A.x = af.x;    A.y = af.y;       // A[m][4s+2h], A[m][4s+2h+1]
      v2f B; B.x = bq[s].x; B.y = bq[s].y;    // B[4s+2h][*], B[4s+2h+1][*]
      acc = __builtin_amdgcn_wmma_f32_16x16x4_f32(false, A, false, B,
                                                  (short)0, acc, false, false);
    }
    // C/D layout: VGPR r = row r (lanes 0-15) / row r+8 (lanes 16-31); cols equal
    if (lane == 0) {
      #pragma unroll
      for (int r = 0; r < 8; ++r) S.sims[tile * 16 + r] = acc[r];
    }
    if (lane == 16) {
      #pragma unroll
      for (int r = 0; r < 8; ++r) S.sims[tile * 16 + 8 + r] = acc[r];
    }
  }
  __syncthreads();

  // ---- 5) cache masked sims per lane (duplicates -> -inf)
  float svv[3];
  for (int t = 0; t < nslots; ++t)
    svv[t] = uf[t] ? S.sims[myJ[t]] : NEG_INF;

  // ---- 6) iterative top-8, ties broken by lower sorted index (== lax.top_k)
  for (int p = 0; p < Kk; ++p) {
    float bv = NEG_INF; int bj = 127;
    for (int t = 0; t < nslots; ++t) {
      bool better = (svv[t] > bv) || (svv[t] == bv && myJ[t] < bj);
      if (better) { bv = svv[t]; bj = myJ[t]; }
    }
    for (int off = 16; off > 0; off >>= 1) {
      float ov = __shfl_xor(bv, off, 32);
      int   oj = __shfl_xor(bj, off, 32);
      bool better = (ov > bv) || (ov == bv && oj < bj);
      if (better) { bv = ov; bj = oj; }
    }
    if (lane == 0) S.topk[p] = S.scand[bj];
    for (int t = 0; t < nslots; ++t)
      if (myJ[t] == bj) svv[t] = NEG_INF;     // remove winner locally
  }

  // ---- 7) fallback: first 8 unique values in ascending order
  {
    int pref = 0, t = 0;
    for (int k = 0; k < P; ++k) {
      if (t < nslots && k == myJ[t]) {
        if (uf[t] && pref < Kk) S.fb[pref] = S.scand[k];
        ++t;
      }
      pref += S.uniqf[k];                      // LDS broadcast
    }
  }
  __syncthreads();

  // ---- 8) emit
  if (lane < Kk && validNode) {
    int res;
    if (num_u >= Kk) {
      res = S.topk[lane];
    } else if (lane < num_u) {
      res = S.fb[lane];
    } else {
      int pi = lane - num_u; if (pi > Kk - 1) pi = Kk - 1;  // clip(pos-num_u,0,K-1)
      res = samples[node * Kk + pi];
    }
    out[node * Kk + lane] = res;
  }
}

extern "C" void kernel_launch(void* const* d_in, const int* in_sizes, int n_in,
                              void* d_out, int out_size, void* d_ws, size_t ws_size,
                              hipStream_t stream) {
  const float* x         = (const float*)d_in[0];  // [N,32] f32
  const int*   neighbors = (const int*)d_in[1];    // [N,8]  i32
  const int*   samples   = (const int*)d_in[2];    // [N,8]  i32
  int*         out       = (int*)d_out;            // [N,8]  i32
  const int N = in_sizes[2] / Kk;
  const int blocks = (N + WPB - 1) / WPB;
  hipLaunchKernelGGL(AttentionSampleUpdater_kernel, dim3(blocks), dim3(256), 0,
                     stream, x, neighbors, samples, out, N);
}